// CrossLayer_47854525612152
// MI455X (gfx1250) — compile-verified
//
#include <hip/hip_runtime.h>

// out[i] = weight[i] * dot(x, x0) + bias[i] + x[i]
// (outer(w,x) @ x0 == w * (x . x0) -- never materialize the 8192^2 matrix:
//  256 MB of traffic vs 160 KB; at 23.3 TB/s that's 11 us vs ~7 ns.)
//
// The dot product is fed through V_WMMA_F32_16X16X4_F32 (exact f32 MAC tree):
//   A[m][k] = x [base + 4m + k]   (16x4)
//   B[k][n] = x0[base + 4n + k]   (4x16)
//   => D[m][m] accumulates the aligned products; trace(D) = 64-elem partial dot.
// Per the CDNA5 A/B VGPR layouts, both operands load from the SAME per-lane
// address: base + 4*(lane&15) + 2*(lane>>4), two consecutive floats (b64 load).

typedef __attribute__((ext_vector_type(2))) float v2f;
typedef __attribute__((ext_vector_type(8))) float v8f;

#define D_TOTAL      8192
#define DOT_BLOCKS   4
#define DOT_THREADS  256
#define WAVES_PER_BLOCK (DOT_THREADS / 32)                       // 8 (wave32)
#define PER_WAVE     (D_TOTAL / (DOT_BLOCKS * WAVES_PER_BLOCK))  // 256
#define WMMA_CHUNK   64
#define DOT_ITERS    (PER_WAVE / WMMA_CHUNK)                     // 4

__global__ __launch_bounds__(DOT_THREADS)
void cross_dot_wmma(const float* __restrict__ x0,
                    const float* __restrict__ x,
                    float* __restrict__ partials) {
  const int lane = threadIdx.x & 31;
  const int wave = threadIdx.x >> 5;
  const int gw   = blockIdx.x * WAVES_PER_BLOCK + wave;
  const int base = gw * PER_WAVE;
  // shared lane->element mapping for the A(16x4) and B(4x16) f32 operands
  const int lsub = (lane & 15) * 4 + (lane >> 4) * 2;

  v8f acc = {0.f, 0.f, 0.f, 0.f, 0.f, 0.f, 0.f, 0.f};
#pragma unroll
  for (int t = 0; t < DOT_ITERS; ++t) {
    const int idx = base + t * WMMA_CHUNK + lsub;
    v2f a = *(const v2f*)(x  + idx);
    v2f b = *(const v2f*)(x0 + idx);
    // 8 args: (neg_a, A, neg_b, B, c_mod, C, reuse_a, reuse_b)
    acc = __builtin_amdgcn_wmma_f32_16x16x4_f32(
        false, a, false, b, (short)0, acc, false, false);
  }

  // trace(D): C/D layout -> vgpr r holds M=r (lanes 0-15, N=lane) and
  // M=r+8 (lanes 16-31, N=lane-16). Diagonal lives at lane r and lane 24+r.
  float diag = 0.f;
  if (lane < 8)        diag = acc[lane];
  else if (lane >= 24) diag = acc[lane - 24];

  // deterministic wave32 xor-tree reduction
#pragma unroll
  for (int off = 16; off > 0; off >>= 1)
    diag += __shfl_xor(diag, off, 32);

  __shared__ float wsum[WAVES_PER_BLOCK];
  if (lane == 0) wsum[wave] = diag;
  __syncthreads();
  if (threadIdx.x == 0) {
    float s = 0.f;
#pragma unroll
    for (int w = 0; w < WAVES_PER_BLOCK; ++w) s += wsum[w];
    partials[blockIdx.x] = s;
  }
}

__global__ __launch_bounds__(256)
void cross_finish(const float* __restrict__ partials,
                  const float* __restrict__ x,
                  const float* __restrict__ weight,
                  const float* __restrict__ bias,
                  float* __restrict__ out) {
  // every block sums the 4 partials in the same fixed order -> deterministic
  float s = 0.f;
#pragma unroll
  for (int b = 0; b < DOT_BLOCKS; ++b) s += partials[b];
  const int i = blockIdx.x * blockDim.x + threadIdx.x;
  if (i < D_TOTAL)
    out[i] = fmaf(weight[i], s, bias[i] + x[i]);
}

extern "C" void kernel_launch(void* const* d_in, const int* in_sizes, int n_in,
                              void* d_out, int out_size, void* d_ws, size_t ws_size,
                              hipStream_t stream) {
  const float* x0     = (const float*)d_in[0];
  const float* x      = (const float*)d_in[1];
  const float* weight = (const float*)d_in[2];
  const float* bias   = (const float*)d_in[3];
  float*       out    = (float*)d_out;
  float*       part   = (float*)d_ws;   // DOT_BLOCKS floats of scratch

  cross_dot_wmma<<<DOT_BLOCKS, DOT_THREADS, 0, stream>>>(x0, x, part);
  cross_finish<<<(D_TOTAL + 255) / 256, 256, 0, stream>>>(part, x, weight, bias, out);
}